// monotonic_binary_output_61555471286543
// MI455X (gfx1250) — compile-verified
//
#include <hip/hip_runtime.h>
#include <hip/hip_bf16.h>

#define LATENT 128
#define ROWSTRIDE (LATENT + 1)   // x rows: 128 feats + z flag
#define BPAD 132                 // 128 + 4 pad -> conflict-free ds_load_b64

typedef __attribute__((ext_vector_type(2))) float v2f;
typedef __attribute__((ext_vector_type(8))) float v8f;

// One wave: 16-row tile of x[:, :128] @ W via V_WMMA_F32_16X16X4_F32 (K=128 in 32 steps).
// A frag: lane L = row L%16, K pair at 4j + 2*(L/16)      -> one global_load_b64 / step
// B frag: lane L = out col L%16 (cols 2..15 zero), same K -> one ds_load_b64 / step,
//         from transposed padded LDS copy of W (no register array, no movrels).
__global__ __launch_bounds__(256) void monotonic_binary_output_kernel(
    const float* __restrict__ x, const float* __restrict__ W,
    const float* __restrict__ bias, const float* __restrict__ u,
    float* __restrict__ out, int n, int ntiles)
{
    __shared__ float Bt[16 * BPAD];      // WMMA B operand, transposed + padded
    __shared__ float prop[8][16][2];     // per-wave logit exchange (128 B / wave)

    const int tid = threadIdx.x;

    // zero-fill (covers pad + the 14 unused columns), then drop in real weights
    for (int i = tid; i < 16 * BPAD; i += 256) Bt[i] = 0.0f;
    __syncthreads();
    {   // 256 threads <-> 2 cols x 128 k
        const int c = tid & 1;
        const int k = tid >> 1;
        Bt[c * BPAD + k] = W[k * 2 + c];
    }
    __syncthreads();

    const int lane = tid & 31;
    const int wave = tid >> 5;
    const int tile = blockIdx.x * 8 + wave;
    if (tile >= ntiles) return;          // uniform per wave -> EXEC all-ones below

    const int col      = lane & 15;
    const int half     = lane >> 4;
    const int row_base = tile * 16;

    // A row fed by this lane; clamp (not predicate) so EXEC stays all-ones for WMMA
    int rowA = row_base + col;
    if (rowA > n - 1) rowA = n - 1;
    const float* xrow = x + (long long)rowA * ROWSTRIDE + 2 * half;
    const float* brow = &Bt[col * BPAD + 2 * half];

    // ---- K loop, fully unrolled: every address is base + immediate offset ----
    v8f acc0 = {}, acc1 = {};
#pragma unroll
    for (int j = 0; j < 32; ++j) {
        const int k = 4 * j;
        v2f a;  a.x  = xrow[k]; a.y  = xrow[k + 1];   // merges to global_load_b64
        v2f bb; bb.x = brow[k]; bb.y = brow[k + 1];   // merges to ds_load_b64
        if (j & 1)
            acc1 = __builtin_amdgcn_wmma_f32_16x16x4_f32(
                       false, a, false, bb, (short)0, acc1, false, false);
        else
            acc0 = __builtin_amdgcn_wmma_f32_16x16x4_f32(
                       false, a, false, bb, (short)0, acc0, false, false);
    }

    // ---- scatter the two useful output columns to LDS ----
    // D layout: lane L, VGPR v -> M = v + 8*(L/16), N = L%16
    if (col < 2) {
        const float bc = bias[col];
#pragma unroll
        for (int v = 0; v < 8; ++v)
            prop[wave][v + 8 * half][col] = acc0[v] + acc1[v] + bc;
    }
    asm volatile("s_wait_dscnt 0" ::: "memory");  // wave-local store->load ordering

    // ---- epilogue: one lane per row ----
    if (lane < 16) {
        const int row = row_base + lane;
        if (row < n) {
            const float t0 = prop[wave][lane][0];
            const float t1 = prop[wave][lane][1];
            const float z  = x[(long long)row * ROWSTRIDE + LATENT];
            // sigmoid monotonic: max/min of probs == sigmoid(max/min of logits)
            const float t  = (z == 1.0f) ? fmaxf(t0, t1) : fminf(t0, t1);
            const float th = 1.0f / (1.0f + __expf(-t));
            out[row] = (u[row] < th) ? 1.0f : 0.0f;
        }
    }
}

extern "C" void kernel_launch(void* const* d_in, const int* in_sizes, int n_in,
                              void* d_out, int out_size, void* d_ws, size_t ws_size,
                              hipStream_t stream) {
    const float* x  = (const float*)d_in[0];   // (N, 129) f32
    const float* W  = (const float*)d_in[1];   // (128, 2) f32
    const float* b  = (const float*)d_in[2];   // (2,)     f32
    const float* u  = (const float*)d_in[3];   // (N,)     f32
    float* out = (float*)d_out;                // (N, 1)   f32

    const int n      = in_sizes[3];            // N = 1,000,000
    const int ntiles = (n + 15) / 16;
    const int blocks = (ntiles + 7) / 8;       // 8 waves (16-row tiles) per block

    monotonic_binary_output_kernel<<<blocks, 256, 0, stream>>>(x, W, b, u, out, n, ntiles);
}